// CLAHE_6416681140825
// MI455X (gfx1250) — compile-verified
//
#include <hip/hip_runtime.h>
#include <hip/hip_bf16.h>

// CLAHE for x: (8,1,2048,2048) fp32, 8x8 tile grid, 256 bins, clip=0.8.
// Phase 1: per-tile histogram + clip/redistribute + WMMA-based 256-prefix-sum -> LUT (d_ws).
// Phase 2: per-pixel bilinear LUT apply (memory-bound: ~400MB traffic, ~17us @ 23.3TB/s).

typedef __attribute__((ext_vector_type(2))) float v2f;
typedef __attribute__((ext_vector_type(8))) float v8f;

#define NB       256
#define TILE     256
#define IMG_W    2048
#define PIXELS   65536      // per tile
#define MAXVAL   204u       // int(0.8*65536//256)

__global__ __launch_bounds__(256) void clahe_hist_lut(const float* __restrict__ x,
                                                      float* __restrict__ lut) {
    __shared__ unsigned int subhist[8][NB];   // per-wave privatized histograms
    __shared__ float hfin[NB];                // clipped+redistributed histogram
    __shared__ float grow[16];                // 16 row sums of the 16x16 view
    __shared__ float scan2[NB];               // fallback scan buffer
    __shared__ unsigned int totalClipped;

    const int t    = threadIdx.x;
    const int wave = t >> 5;
    const int tile = blockIdx.x;              // 0..511 = b*64 + gy*8 + gx
    const int b    = tile >> 6;
    const int gy   = (tile >> 3) & 7;
    const int gx   = tile & 7;

    #pragma unroll
    for (int w = 0; w < 8; ++w) subhist[w][t] = 0u;
    if (t == 0) totalClipped = 0u;
    __syncthreads();

    // ---- histogram: 64 iters, each covers 4 rows (256 thr * float4) ----
    const float* base = x + ((size_t)b << 22) + (size_t)(gy * TILE) * IMG_W + gx * TILE;
    unsigned int* myhist = subhist[wave];
    const int rsub = t >> 6;                  // 0..3
    const int col  = (t & 63) << 2;           // 0..252
    for (int i = 0; i < 64; ++i) {
        const int row = i * 4 + rsub;
        const float4 v = *reinterpret_cast<const float4*>(base + (size_t)row * IMG_W + col);
        int b0 = min(max((int)(v.x * 256.0f), 0), 255);
        int b1 = min(max((int)(v.y * 256.0f), 0), 255);
        int b2 = min(max((int)(v.z * 256.0f), 0), 255);
        int b3 = min(max((int)(v.w * 256.0f), 0), 255);
        atomicAdd(&myhist[b0], 1u);
        atomicAdd(&myhist[b1], 1u);
        atomicAdd(&myhist[b2], 1u);
        atomicAdd(&myhist[b3], 1u);
    }
    __syncthreads();

    // ---- merge sub-histograms, clip, sum ----
    unsigned int h = 0;
    #pragma unroll
    for (int w = 0; w < 8; ++w) h += subhist[w][t];
    unsigned int hc = (h > MAXVAL) ? MAXVAL : h;
    atomicAdd(&totalClipped, hc);
    __syncthreads();

    // ---- redistribute ----
    const unsigned int clipped  = (unsigned int)PIXELS - totalClipped;
    const unsigned int residual = clipped & 255u;
    const unsigned int redist   = clipped >> 8;
    hfin[t] = (float)(hc + redist + (((unsigned)t < residual) ? 1u : 0u));
    __syncthreads();

    // row sums of 16x16 view (for the L x (H x J) term)
    if (t < 16) {
        float s = 0.0f;
        #pragma unroll
        for (int c = 0; c < 16; ++c) s += hfin[t * 16 + c];
        grow[t] = s;
    }
    __syncthreads();

    const float scale = 255.0f / 65536.0f;

#if __has_builtin(__builtin_amdgcn_wmma_f32_16x16x4_f32)
    // ---- 256-bin inclusive prefix sum via WMMA on wave 0 (EXEC all-ones) ----
    // H viewed as 16x16 row-major: P = H*U + Lstrict*G, G[k][n] = grow[k].
    if (t < 32) {
        const int half = t >> 4;     // 0: K=0,1 half | 1: K=2,3 half (A/B frags)
        const int ln   = t & 15;     // M (for A) / N (for B)
        v8f acc = {};
        #pragma unroll
        for (int kb = 0; kb < 4; ++kb) {        // P += H(:,4kb..) * U(4kb..,:)
            const int k0 = kb * 4 + 2 * half;
            v2f a, bb;
            a.x  = hfin[ln * 16 + k0];
            a.y  = hfin[ln * 16 + k0 + 1];
            bb.x = (k0     <= ln) ? 1.0f : 0.0f;   // U[k][n] = (k<=n)
            bb.y = (k0 + 1 <= ln) ? 1.0f : 0.0f;
            acc = __builtin_amdgcn_wmma_f32_16x16x4_f32(
                false, a, false, bb, (short)0, acc, false, false);
        }
        #pragma unroll
        for (int kb = 0; kb < 4; ++kb) {        // P += Lstrict(:,4kb..) * G(4kb..,:)
            const int k0 = kb * 4 + 2 * half;
            v2f a, bb;
            a.x  = (k0     < ln) ? 1.0f : 0.0f;    // Lstrict[m][k] = (k<m)
            a.y  = (k0 + 1 < ln) ? 1.0f : 0.0f;
            bb.x = grow[k0];
            bb.y = grow[k0 + 1];
            acc = __builtin_amdgcn_wmma_f32_16x16x4_f32(
                false, a, false, bb, (short)0, acc, false, false);
        }
        // D layout: vgpr v, lanes<16 -> M=v ; lanes>=16 -> M=v+8 ; N = ln ; bin = 16M+N
        float* lt = lut + (size_t)tile * NB;
        #pragma unroll
        for (int v = 0; v < 8; ++v) {
            const int bin = 16 * (v + 8 * half) + ln;
            float p = acc[v] * scale;
            lt[bin] = floorf(fminf(fmaxf(p, 0.0f), 255.0f));
        }
    }
#else
    // ---- fallback: serial scan (tiny; 512 blocks) ----
    if (t == 0) {
        float s = 0.0f;
        for (int i = 0; i < NB; ++i) { s += hfin[i]; scan2[i] = s; }
    }
    __syncthreads();
    lut[(size_t)tile * NB + t] = floorf(fminf(fmaxf(scan2[t] * scale, 0.0f), 255.0f));
#endif
}

__global__ __launch_bounds__(256) void clahe_apply(const float* __restrict__ x,
                                                   const float* __restrict__ lut,
                                                   float* __restrict__ out) {
    const size_t pix = ((size_t)blockIdx.x * 256 + threadIdx.x) * 4;  // 4 px / thread
    const int b   = (int)(pix >> 22);
    const int rem = (int)(pix & ((1u << 22) - 1));
    const int y   = rem >> 11;
    const int xc  = rem & 2047;

    const float yy  = (y + 0.5f) * (1.0f / 256.0f) - 0.5f;
    const float y0f = floorf(yy);
    const int y0i = (int)y0f;
    const int y0  = min(max(y0i, 0), 7);
    const int y1  = min(max(y0i + 1, 0), 7);
    const float wy = yy - y0f;

    const float4 v = *reinterpret_cast<const float4*>(x + pix);
    const float* lb = lut + (size_t)b * 64 * NB;
    float4 o;
    const float* vp = &v.x;
    float* op = &o.x;

    #pragma unroll
    for (int j = 0; j < 4; ++j) {
        const int xj = xc + j;
        const float xx  = (xj + 0.5f) * (1.0f / 256.0f) - 0.5f;
        const float x0f = floorf(xx);
        const int x0i = (int)x0f;
        const int x0  = min(max(x0i, 0), 7);
        const int x1  = min(max(x0i + 1, 0), 7);
        const float wx = xx - x0f;

        const float val = vp[j];
        const int bidx = min(max((int)(val * 255.0f), 0), 255);

        const float v00 = lb[(y0 * 8 + x0) * NB + bidx];
        const float v01 = lb[(y0 * 8 + x1) * NB + bidx];
        const float v10 = lb[(y1 * 8 + x0) * NB + bidx];
        const float v11 = lb[(y1 * 8 + x1) * NB + bidx];

        const float r = (1.0f - wy) * ((1.0f - wx) * v00 + wx * v01)
                      +         wy  * ((1.0f - wx) * v10 + wx * v11);
        op[j] = r / 255.0f;
    }
    *reinterpret_cast<float4*>(out + pix) = o;
}

extern "C" void kernel_launch(void* const* d_in, const int* in_sizes, int n_in,
                              void* d_out, int out_size, void* d_ws, size_t ws_size,
                              hipStream_t stream) {
    const float* x = (const float*)d_in[0];
    float* out = (float*)d_out;
    float* lut = (float*)d_ws;   // 512 tiles * 256 bins * 4B = 512 KB

    clahe_hist_lut<<<512, 256, 0, stream>>>(x, lut);
    // 8*2048*2048 px / (256 thr * 4 px) = 32768 blocks
    clahe_apply<<<32768, 256, 0, stream>>>(x, lut, out);
}